// GHMR_10273561772277
// MI455X (gfx1250) — compile-verified
//
#include <hip/hip_runtime.h>

typedef float f4  __attribute__((ext_vector_type(4)));
typedef float v2f __attribute__((ext_vector_type(2)));
typedef float v8f __attribute__((ext_vector_type(8)));

#define MU_F   0.02f
#define NBINS  10
#define NQ     21        // [bin0 cnt, bin0 sum, ..., bin9 cnt, bin9 sum, totw]
#define TPB    256
#define PAD    22        // dwords per per-thread LDS slot (20 used; 22 => conflict-free banks)
#define MAXBLK 1024

// xor-butterfly step via ds_swizzle (group-of-32: and=0x1F, or=0, xor=mask)
#define SWZ_XOR_ADD(s, mask)                                                    \
    do {                                                                        \
        int _r = __builtin_amdgcn_ds_swizzle(__builtin_bit_cast(int, (s)),      \
                                             (0x1F | ((mask) << 10)));          \
        (s) += __builtin_bit_cast(float, _r);                                   \
    } while (0)

// Wave32 sum-reduction through the CDNA5 matrix pipe.
// A = ones(16x4)  =>  D[m][n] = sum_k B[k][n].  Lane partials go into B
// (VGPR1 = 0); every physical B slot maps to a unique (k,n), so summing all 16
// column sums (D VGPR0: lane l holds colsum[l&15] per the documented C/D
// layout) yields the exact 32-lane total independent of B's slot mapping.
__device__ __forceinline__ float wave_reduce_sum(float v) {
    v2f a; a[0] = 1.0f; a[1] = 1.0f;
    v2f b; b[0] = v;    b[1] = 0.0f;
    v8f c = {};
    c = __builtin_amdgcn_wmma_f32_16x16x4_f32(false, a, false, b,
                                              (short)0, c, false, false);
    float s = c[0];                 // lane l: colsum[l & 15]
    SWZ_XOR_ADD(s, 1);
    SWZ_XOR_ADD(s, 2);
    SWZ_XOR_ADD(s, 4);
    SWZ_XOR_ADD(s, 8);
    return s;                       // all lanes hold the wave total
}

__device__ __forceinline__ void ghmr_elem(float p, float t, float w,
                                          float* slot, float& totw) {
    const float mu2 = MU_F * MU_F;
    float d    = p - t;
    float x    = __builtin_fmaf(d, d, mu2);      // d^2 + mu^2  (>= mu^2, no denorm)
    float r    = __builtin_amdgcn_rsqf(x);       // v_rsq_f32, 1 TRANS op
    float s    = x * r;                          // sqrt(x) to ~2 ulp
    float loss = s - MU_F;
    float t10  = __builtin_fabsf(d) * r * 10.0f; // 10 * g, g in [0,1)
    int bin    = (int)t10;                       // trunc == floor (t10 >= 0)
    bin        = bin > (NBINS - 1) ? (NBINS - 1) : bin;
    float cv   = (w > 0.0f) ? 1.0f : 0.0f;
    float lw   = loss * __builtin_fmaxf(w, 0.0f);  // 0 when invalid (loss >= 0)
    totw += w;
    // fire-and-forget ds_add_f32: no VGPR dependency -> no dscnt stall in loop;
    // private per-thread slot -> no contention.
    atomicAdd(&slot[bin * 2],     cv);
    atomicAdd(&slot[bin * 2 + 1], lw);
}

__global__ __launch_bounds__(TPB) void ghmr_pass1(
    const f4* __restrict__ pred4, const f4* __restrict__ targ4,
    const f4* __restrict__ wgt4,
    const float* __restrict__ pred, const float* __restrict__ targ,
    const float* __restrict__ wgt,
    float* __restrict__ partials, int nvec, int ntot)
{
    __shared__ float hist[TPB * PAD];
    __shared__ float wavebuf[(TPB / 32) * NQ];

    const int tid = threadIdx.x;
    float* mySlot = &hist[tid * PAD];

    #pragma unroll
    for (int q = 0; q < 2 * NBINS; ++q) mySlot[q] = 0.0f;

    float totw = 0.0f;
    const int stride = gridDim.x * TPB;

    for (int i = blockIdx.x * TPB + tid; i < nvec; i += stride) {
        f4 p = pred4[i];                 // plain RT loads: let L2 keep the
        f4 t = targ4[i];                 // 192MB working set across replays
        f4 w = wgt4[i];
        #pragma unroll
        for (int k = 0; k < 4; ++k)
            ghmr_elem(p[k], t[k], w[k], mySlot, totw);
    }

    // scalar tail (ntot not multiple of 4) -- one thread
    if (blockIdx.x == 0 && tid == 0) {
        for (int i = nvec * 4; i < ntot; ++i)
            ghmr_elem(pred[i], targ[i], wgt[i], mySlot, totw);
    }

    // Wave-level reduction of the 21 quantities via WMMA (uniform control
    // flow across the block here => EXEC all ones for v_wmma).  DS ops are
    // in-order per wave, so the readback sees all prior ds_add_f32.
    const int wave = tid >> 5;
    const int lane = tid & 31;
    for (int q = 0; q < NQ; ++q) {
        float v = (q < 2 * NBINS) ? mySlot[q] : totw;
        float s = wave_reduce_sum(v);
        if (lane == 0) wavebuf[wave * NQ + q] = s;
    }
    __syncthreads();

    // Cross-wave (8 partials, fixed order) -> per-block partials in d_ws.
    if (tid < NQ) {
        float acc = 0.0f;
        #pragma unroll
        for (int wv = 0; wv < TPB / 32; ++wv) acc += wavebuf[wv * NQ + tid];
        partials[blockIdx.x * NQ + tid] = acc;   // plain store: overwrites poison
    }
}

__global__ __launch_bounds__(64) void ghmr_finalize(
    const float* __restrict__ partials, float* __restrict__ out,
    int nblocks, int ntot)
{
    __shared__ double vals[NQ];
    const int tid = threadIdx.x;
    if (tid < NQ) {
        double acc = 0.0;                        // fixed-order, deterministic
        for (int b = 0; b < nblocks; ++b) acc += (double)partials[b * NQ + tid];
        vals[tid] = acc;
    }
    __syncthreads();
    if (tid == 0) {
        double tot = vals[20];
        if (tot < 1.0) tot = 1.0;
        double nn = 0.0, acc = 0.0;
        for (int b = 0; b < NBINS; ++b) {
            double cnt = vals[b * 2];
            double S   = vals[b * 2 + 1];
            if (cnt > 0.0) {
                nn += 1.0;
                double c1 = cnt < 1.0 ? 1.0 : cnt;
                acc += (tot / c1) * S;
            }
        }
        double denom = nn < 1.0 ? 1.0 : nn;
        acc = acc / denom / (double)ntot;        // mean over all N*D elements
        out[0] = (float)acc;
    }
}

extern "C" void kernel_launch(void* const* d_in, const int* in_sizes, int n_in,
                              void* d_out, int out_size, void* d_ws, size_t ws_size,
                              hipStream_t stream)
{
    (void)n_in; (void)out_size;
    const float* pred = (const float*)d_in[0];
    const float* targ = (const float*)d_in[1];
    const float* wgt  = (const float*)d_in[2];
    const int ntot = in_sizes[0];        // N*D = 16,777,216
    const int nvec = ntot / 4;

    float* partials = (float*)d_ws;
    size_t maxb = ws_size / (NQ * sizeof(float));
    int nblocks = MAXBLK;
    if ((size_t)nblocks > maxb) nblocks = (int)maxb;
    if (nblocks < 1) nblocks = 1;

    ghmr_pass1<<<nblocks, TPB, 0, stream>>>(
        (const f4*)pred, (const f4*)targ, (const f4*)wgt,
        pred, targ, wgt, partials, nvec, ntot);
    ghmr_finalize<<<1, 64, 0, stream>>>(partials, (float*)d_out, nblocks, ntot);
}